// GAT_82334523064895
// MI455X (gfx1250) — compile-verified
//
#include <hip/hip_runtime.h>
#include <math.h>

// ---------------- problem constants (match reference) ----------------
#define NODES 8192
#define NEDGE 262144
#define FIN   8192
#define HDIM  16
#define CDIM  2
#define ETOT  (NEDGE + NODES)   // edges + self loops = 270336
#define NEG_SLOPE 0.2f
#define KSPLIT 8                // K-split ways for GEMM1 (occupancy)

typedef __attribute__((ext_vector_type(2))) float v2f;
typedef __attribute__((ext_vector_type(8))) float v8f;

// ---------------- float atomic min/max via monotone int/uint trick ----
__device__ __forceinline__ void atomicMaxF(float* a, float v) {
  if (v >= 0.0f) atomicMax((int*)a, __float_as_int(v));
  else           atomicMin((unsigned int*)a, __float_as_uint(v));
}
__device__ __forceinline__ void atomicMinF(float* a, float v) {
  if (v >= 0.0f) atomicMin((int*)a, __float_as_int(v));
  else           atomicMax((unsigned int*)a, __float_as_uint(v));
}

// ---------------- generic fill -----------------------------------------
__global__ void fill_f32(float* __restrict__ p, float v, int n) {
  int i = blockIdx.x * blockDim.x + threadIdx.x;
  if (i < n) p[i] = v;
}

// ---------------- GEMM1: h = x @ W1 via V_WMMA_F32_16X16X4_F32 ---------
// One wave computes a 16x16 f32 tile over a K-slice of 1024, 4 independent
// accumulators, A streamed with non-temporal b64 loads, partials reduced
// with global f32 atomics into h (8192x16).
__global__ __launch_bounds__(256) void gemm1_wmma(const float* __restrict__ x,
                                                  const float* __restrict__ W1,
                                                  float* __restrict__ h1) {
  const int wid     = (int)((blockIdx.x * blockDim.x + threadIdx.x) >> 5);
  const int lane    = threadIdx.x & 31;
  const int rowTile = wid / KSPLIT;            // 0..511
  const int kslice  = wid % KSPLIT;            // 0..7
  const int row0    = rowTile * 16;
  const int klen    = FIN / KSPLIT;            // 1024
  const int kbase   = kslice * klen;
  const int m       = lane & 15;               // M for A, N for B/C
  const int kh      = (lane >> 4) << 1;        // 0 (lanes 0-15) or 2 (lanes 16-31)

  // A[m][k] = x[(row0+m)*FIN + kbase + k]; lane holds K pair (kh, kh+1) -> b64
  const float* ap = x  + (size_t)(row0 + m) * FIN + (size_t)(kbase + kh);
  // B[k][n] = W1[(kbase+k)*HDIM + n]; lane n=m holds rows kh, kh+1
  const float* bp = W1 + (size_t)(kbase + kh) * HDIM + m;

  v8f acc0 = {}; v8f acc1 = {}; v8f acc2 = {}; v8f acc3 = {};
  for (int kk = 0; kk < klen; kk += 16) {
    v2f a0 = __builtin_nontemporal_load((const v2f*)(ap + kk));
    v2f a1 = __builtin_nontemporal_load((const v2f*)(ap + kk + 4));
    v2f a2 = __builtin_nontemporal_load((const v2f*)(ap + kk + 8));
    v2f a3 = __builtin_nontemporal_load((const v2f*)(ap + kk + 12));
    v2f b0, b1, b2, b3;
    b0.x = bp[(size_t)(kk     ) * HDIM]; b0.y = bp[(size_t)(kk     ) * HDIM + HDIM];
    b1.x = bp[(size_t)(kk +  4) * HDIM]; b1.y = bp[(size_t)(kk +  4) * HDIM + HDIM];
    b2.x = bp[(size_t)(kk +  8) * HDIM]; b2.y = bp[(size_t)(kk +  8) * HDIM + HDIM];
    b3.x = bp[(size_t)(kk + 12) * HDIM]; b3.y = bp[(size_t)(kk + 12) * HDIM + HDIM];
    acc0 = __builtin_amdgcn_wmma_f32_16x16x4_f32(false, a0, false, b0, (short)0, acc0, false, false);
    acc1 = __builtin_amdgcn_wmma_f32_16x16x4_f32(false, a1, false, b1, (short)0, acc1, false, false);
    acc2 = __builtin_amdgcn_wmma_f32_16x16x4_f32(false, a2, false, b2, (short)0, acc2, false, false);
    acc3 = __builtin_amdgcn_wmma_f32_16x16x4_f32(false, a3, false, b3, (short)0, acc3, false, false);
  }
  v8f acc = (acc0 + acc1) + (acc2 + acc3);

  // C/D layout: VGPR i holds row (i + 8*(lane>>4)), col = lane&15
  const int mo = 8 * (lane >> 4);
#pragma unroll
  for (int i = 0; i < 8; ++i)
    atomicAdd(&h1[(size_t)(row0 + mo + i) * HDIM + m], acc[i]);
}

// ---------------- per-node attention scores ----------------------------
__global__ void node_scores(const float* __restrict__ h, const float* __restrict__ asrc,
                            const float* __restrict__ adst, float* __restrict__ ssrc,
                            float* __restrict__ sdst, int F) {
  int i = blockIdx.x * blockDim.x + threadIdx.x;
  if (i >= NODES) return;
  float s = 0.0f, d = 0.0f;
  for (int f = 0; f < F; ++f) {
    float hv = h[(size_t)i * F + f];
    s += hv * asrc[f];
    d += hv * adst[f];
  }
  ssrc[i] = s;
  sdst[i] = d;
}

// ---------------- edge pass 1: e = LeakyReLU, segment max over dst -----
__global__ void edge_max(const int* __restrict__ esrc, const int* __restrict__ edst,
                         const float* __restrict__ ssrc, const float* __restrict__ sdst,
                         float* __restrict__ ebuf, float* __restrict__ mx) {
  int i = blockIdx.x * blockDim.x + threadIdx.x;
  if (i >= ETOT) return;
  int s, d;
  if (i < NEDGE) { s = esrc[i]; d = edst[i]; }
  else           { s = d = i - NEDGE; }           // self loops appended
  float e = ssrc[s] + sdst[d];
  e = (e > 0.0f) ? e : NEG_SLOPE * e;
  ebuf[i] = e;
  atomicMaxF(&mx[d], e);
}

// ---------------- edge pass 2: exp, denom, weighted aggregate ----------
// One thread per (edge, feature). sum(ex*h)/sum(ex) == sum(alpha*h).
__global__ void edge_accum(const int* __restrict__ esrc, const int* __restrict__ edst,
                           const float* __restrict__ ebuf, const float* __restrict__ mx,
                           const float* __restrict__ h, float* __restrict__ denom,
                           float* __restrict__ agg, int F) {
  long long t = (long long)blockIdx.x * blockDim.x + threadIdx.x;
  if (t >= (long long)ETOT * F) return;
  int i = (int)(t / F);
  int f = (int)(t % F);
  int s, d;
  if (i < NEDGE) { s = esrc[i]; d = edst[i]; }
  else           { s = d = i - NEDGE; }
  float ex = expf(ebuf[i] - mx[d]);
  if (f == 0) atomicAdd(&denom[d], ex);
  atomicAdd(&agg[(size_t)d * F + f], ex * h[(size_t)s * F + f]);
}

// ---------------- finalize layer: divide by denom, add bias ------------
__global__ void finalize_agg(float* __restrict__ agg, const float* __restrict__ denom,
                             const float* __restrict__ bias, int F) {
  int t = blockIdx.x * blockDim.x + threadIdx.x;
  if (t >= NODES * F) return;
  int i = t / F, f = t % F;
  agg[t] = agg[t] / (denom[i] + 1e-16f) + bias[f];
}

// ---------------- layer2 node work: h2 = x2 @ W2 + scores --------------
__global__ void layer2_node(const float* __restrict__ x2, const float* __restrict__ W2,
                            const float* __restrict__ as2, const float* __restrict__ ad2,
                            float* __restrict__ h2, float* __restrict__ ssrc,
                            float* __restrict__ sdst) {
  int i = blockIdx.x * blockDim.x + threadIdx.x;
  if (i >= NODES) return;
  float c0 = 0.0f, c1 = 0.0f;
#pragma unroll
  for (int f = 0; f < HDIM; ++f) {
    float v = x2[(size_t)i * HDIM + f];
    c0 += v * W2[f * CDIM + 0];
    c1 += v * W2[f * CDIM + 1];
  }
  h2[(size_t)i * CDIM + 0] = c0;
  h2[(size_t)i * CDIM + 1] = c1;
  ssrc[i] = c0 * as2[0] + c1 * as2[1];
  sdst[i] = c0 * ad2[0] + c1 * ad2[1];
}

// ---------------- finalize layer2 + global min/max ---------------------
__global__ void finalize_minmax(float* __restrict__ agg, const float* __restrict__ denom,
                                const float* __restrict__ bias, float* __restrict__ mnmx) {
  int t = blockIdx.x * blockDim.x + threadIdx.x;
  if (t >= NODES * CDIM) return;
  int i = t / CDIM, c = t % CDIM;
  float v = agg[t] / (denom[i] + 1e-16f) + bias[c];
  agg[t] = v;
  atomicMinF(&mnmx[0], v);
  atomicMaxF(&mnmx[1], v);
}

// ---------------- rescale to [-1, 1] -----------------------------------
__global__ void rescale(const float* __restrict__ v, const float* __restrict__ mnmx,
                        float* __restrict__ out) {
  int t = blockIdx.x * blockDim.x + threadIdx.x;
  if (t >= NODES * CDIM) return;
  float mn = mnmx[0], mx = mnmx[1];
  out[t] = 2.0f * ((v[t] - mn) / (mx - mn)) - 1.0f;
}

// ---------------- workspace layout (float offsets) ---------------------
#define WS_H1     0                         // 8192*16
#define WS_X2     (WS_H1 + NODES*HDIM)      // 8192*16 (agg1 -> x2 in place)
#define WS_SSRC1  (WS_X2 + NODES*HDIM)      // 8192
#define WS_SDST1  (WS_SSRC1 + NODES)        // 8192
#define WS_M1     (WS_SDST1 + NODES)        // 8192
#define WS_DEN1   (WS_M1 + NODES)           // 8192
#define WS_E1     (WS_DEN1 + NODES)         // 270336
#define WS_H2     (WS_E1 + ETOT)            // 8192*2
#define WS_SSRC2  (WS_H2 + NODES*CDIM)      // 8192
#define WS_SDST2  (WS_SSRC2 + NODES)        // 8192
#define WS_M2     (WS_SDST2 + NODES)        // 8192
#define WS_DEN2   (WS_M2 + NODES)           // 8192
#define WS_E2     (WS_DEN2 + NODES)         // 270336
#define WS_AGG2   (WS_E2 + ETOT)            // 8192*2
#define WS_MNMX   (WS_AGG2 + NODES*CDIM)    // 2

extern "C" void kernel_launch(void* const* d_in, const int* in_sizes, int n_in,
                              void* d_out, int out_size, void* d_ws, size_t ws_size,
                              hipStream_t stream) {
  const float* x     = (const float*)d_in[0];
  const int*   ei    = (const int*)d_in[1];   // [2, E] flat: src then dst
  const float* W1    = (const float*)d_in[2];
  const float* asrc1 = (const float*)d_in[3];
  const float* adst1 = (const float*)d_in[4];
  const float* b1    = (const float*)d_in[5];
  const float* W2    = (const float*)d_in[6];
  const float* asrc2 = (const float*)d_in[7];
  const float* adst2 = (const float*)d_in[8];
  const float* b2    = (const float*)d_in[9];
  float* out = (float*)d_out;
  float* ws  = (float*)d_ws;

  const int* esrc = ei;
  const int* edst = ei + NEDGE;

  const float NINF = -__builtin_inff();
  const float PINF =  __builtin_inff();

  // ---- init (atomically-accumulated buffers must be re-zeroed each call)
  fill_f32<<<(NODES*HDIM + 255)/256, 256, 0, stream>>>(ws + WS_H1,   0.0f, NODES*HDIM);
  fill_f32<<<(NODES*HDIM + 255)/256, 256, 0, stream>>>(ws + WS_X2,   0.0f, NODES*HDIM);
  fill_f32<<<(NODES + 255)/256,      256, 0, stream>>>(ws + WS_M1,   NINF, NODES);
  fill_f32<<<(NODES + 255)/256,      256, 0, stream>>>(ws + WS_DEN1, 0.0f, NODES);
  fill_f32<<<(NODES + 255)/256,      256, 0, stream>>>(ws + WS_M2,   NINF, NODES);
  fill_f32<<<(NODES + 255)/256,      256, 0, stream>>>(ws + WS_DEN2, 0.0f, NODES);
  fill_f32<<<(NODES*CDIM + 255)/256, 256, 0, stream>>>(ws + WS_AGG2, 0.0f, NODES*CDIM);
  fill_f32<<<1, 32, 0, stream>>>(ws + WS_MNMX,     PINF, 1);
  fill_f32<<<1, 32, 0, stream>>>(ws + WS_MNMX + 1, NINF, 1);

  // ---- layer 1: h = x @ W1 (WMMA f32, K-split 8, 4096 waves)
  gemm1_wmma<<<(NODES/16) * KSPLIT * 32 / 256, 256, 0, stream>>>(x, W1, ws + WS_H1);

  // ---- layer 1 attention
  node_scores<<<(NODES + 255)/256, 256, 0, stream>>>(ws + WS_H1, asrc1, adst1,
                                                     ws + WS_SSRC1, ws + WS_SDST1, HDIM);
  edge_max<<<(ETOT + 255)/256, 256, 0, stream>>>(esrc, edst, ws + WS_SSRC1, ws + WS_SDST1,
                                                 ws + WS_E1, ws + WS_M1);
  {
    long long tot = (long long)ETOT * HDIM;
    edge_accum<<<(int)((tot + 255)/256), 256, 0, stream>>>(esrc, edst, ws + WS_E1, ws + WS_M1,
                                                           ws + WS_H1, ws + WS_DEN1,
                                                           ws + WS_X2, HDIM);
  }
  finalize_agg<<<(NODES*HDIM + 255)/256, 256, 0, stream>>>(ws + WS_X2, ws + WS_DEN1, b1, HDIM);

  // ---- layer 2
  layer2_node<<<(NODES + 255)/256, 256, 0, stream>>>(ws + WS_X2, W2, asrc2, adst2,
                                                     ws + WS_H2, ws + WS_SSRC2, ws + WS_SDST2);
  edge_max<<<(ETOT + 255)/256, 256, 0, stream>>>(esrc, edst, ws + WS_SSRC2, ws + WS_SDST2,
                                                 ws + WS_E2, ws + WS_M2);
  {
    long long tot = (long long)ETOT * CDIM;
    edge_accum<<<(int)((tot + 255)/256), 256, 0, stream>>>(esrc, edst, ws + WS_E2, ws + WS_M2,
                                                           ws + WS_H2, ws + WS_DEN2,
                                                           ws + WS_AGG2, CDIM);
  }
  finalize_minmax<<<(NODES*CDIM + 255)/256, 256, 0, stream>>>(ws + WS_AGG2, ws + WS_DEN2, b2,
                                                              ws + WS_MNMX);
  rescale<<<(NODES*CDIM + 255)/256, 256, 0, stream>>>(ws + WS_AGG2, ws + WS_MNMX, out);
}